// PathExplosion_17231408792016
// MI455X (gfx1250) — compile-verified
//
#include <hip/hip_runtime.h>
#include <hip/hip_bf16.h>
#include <math.h>

// PathExplosion on MI455X (gfx1250, wave32):
//   Phase 1: bound[b] = sigmoid(relu(x@w1+b1)@w2+b2) via V_WMMA_F32_16X16X4_F32
//            (f32 WMMA: the count output is branch-sensitive to 'bound', so no
//             low-precision matrix path is admissible).
//   Phase 2: exact per-element replay of the while-loop (pure f32 VALU,
//            latency-bound; memory traffic is negligible at 23.3 TB/s).

typedef __attribute__((ext_vector_type(2))) float v2f;
typedef __attribute__((ext_vector_type(8))) float v8f;

// ---------------------------------------------------------------------------
// Kernel 1: one wave computes bound[] for 16 rows.
//
// z[m] = sum_{j=0..63} relu(x0[m]*w1[0,j] + x1[m]*w1[1,j] + b1[j]) * w2[j]
//
// K=64 is split into 16 chunks of 4 and fed to V_WMMA_F32_16X16X4_F32 with
// B's 16 columns all replicated to w2[k..k+3], so D[m][n] == z[m] for all n.
//
// A-matrix 16x4 f32 layout (ISA 7.12.2): lane L holds row m = L&15,
//   VGPR0/VGPR1 = K = halfK / halfK+1 where halfK = (L>>4)*2.
// B-matrix 4x16 mirrors this striping, and since our B depends only on K,
// lane L loads w2 at exactly the same j indices as its A elements.
// D layout: lane n<16 holds D[v][n] in acc[v] (rows 0..7);
//           lane n>=16 holds D[8+v][n-16] (rows 8..15).
// ---------------------------------------------------------------------------
__global__ __launch_bounds__(256) void pe_bound_kernel(
    const float* __restrict__ x,   // [B,2]
    const float* __restrict__ w1,  // [2,64] row-major
    const float* __restrict__ b1,  // [64]
    const float* __restrict__ w2,  // [64]
    const float* __restrict__ b2,  // [1]
    float* __restrict__ bound,     // [B]  (workspace)
    int Btot)
{
  const int lane = threadIdx.x & 31;
  const int wave = threadIdx.x >> 5;
  const int rowTile = (blockIdx.x * 8 + wave) * 16;   // 16 rows per wave
  if (rowTile >= Btot) return;                        // wave-uniform guard

  const int m     = lane & 15;
  const int halfK = (lane >> 4) << 1;                 // 0 or 2
  const int row   = rowTile + m;
  const float x0  = x[row * 2 + 0];
  const float x1  = x[row * 2 + 1];

  v8f acc = {};
#pragma unroll
  for (int c = 0; c < 16; ++c) {
    const int j0 = c * 4 + halfK;
    // Layer 1 (K=2, done in VALU): two h values this lane owns.
    float h0 = fmaf(x1, w1[64 + j0    ], fmaf(x0, w1[j0    ], b1[j0    ]));
    float h1 = fmaf(x1, w1[64 + j0 + 1], fmaf(x0, w1[j0 + 1], b1[j0 + 1]));
    h0 = h0 > 0.0f ? h0 : 0.0f;
    h1 = h1 > 0.0f ? h1 : 0.0f;

    v2f a; a.x = h0;        a.y = h1;
    v2f b; b.x = w2[j0];    b.y = w2[j0 + 1];

    // D = A*B + C  (f32, 16x16x4)
    acc = __builtin_amdgcn_wmma_f32_16x16x4_f32(
        /*neg_a=*/false, a, /*neg_b=*/false, b,
        /*c_mod=*/(short)0, acc, /*reuse_a=*/false, /*reuse_b=*/false);
  }

  const float bias2 = b2[0];
  // Column 0 of D lives in lane 0 (rows 0..7) and lane 16 (rows 8..15).
  if (lane == 0 || lane == 16) {
    const int base = rowTile + ((lane >> 4) << 3);
#pragma unroll
    for (int v = 0; v < 8; ++v) {
      const float z = acc[v] + bias2;
      bound[base + v] = 1.0f / (1.0f + expf(-z));     // sigmoid
    }
  }
}

// ---------------------------------------------------------------------------
// Kernel 2: exact replay of the reference while-loop, one thread per element.
// Inactive elements are fixed points in the reference, so per-element
// termination is semantically identical to the global lax.while_loop.
// ---------------------------------------------------------------------------
__global__ __launch_bounds__(256) void pe_count_kernel(
    const float* __restrict__ x,      // [B,2]
    const float* __restrict__ bound,  // [B]
    float* __restrict__ out,          // [B,2]
    int n)                            // B*2
{
  const int i = blockIdx.x * blockDim.x + threadIdx.x;
  if (i >= n) return;

  const float bd = bound[i >> 1];
  const float bm = bd - 0.001f;       // bm <= bd, masks are nested
  float xc = x[i];
  float c  = 0.0f;

  while (xc <= 10.0f) {
    xc += 0.01f;
    if (xc <= bm)       xc *= 2.0f;   // m_in & m_lo
    else if (xc <= bd)  xc *= 3.0f;   // m_in & ~m_lo
    c += 1.0f;
  }
  out[i] = c;
}

// ---------------------------------------------------------------------------
extern "C" void kernel_launch(void* const* d_in, const int* in_sizes, int n_in,
                              void* d_out, int out_size, void* d_ws, size_t ws_size,
                              hipStream_t stream) {
  const float* x  = (const float*)d_in[0];   // [B,2]
  const float* w1 = (const float*)d_in[1];   // [2,64]
  const float* b1 = (const float*)d_in[2];   // [64]
  const float* w2 = (const float*)d_in[3];   // [64,1]
  const float* b2 = (const float*)d_in[4];   // [1]
  float* out = (float*)d_out;

  const int Btot = in_sizes[0] / 2;          // 65536
  float* bound = (float*)d_ws;               // Btot floats = 256 KB scratch

  // Phase 1: 8 waves/block * 16 rows/wave = 128 rows per block.
  const int rowsPerBlock = 128;
  dim3 g1((Btot + rowsPerBlock - 1) / rowsPerBlock);
  pe_bound_kernel<<<g1, 256, 0, stream>>>(x, w1, b1, w2, b2, bound, Btot);

  // Phase 2: one thread per output element.
  const int n = Btot * 2;
  dim3 g2((n + 255) / 256);
  pe_count_kernel<<<g2, 256, 0, stream>>>(x, bound, out, n);
}